// SegmenterTensorFlow_91293824843826
// MI455X (gfx1250) — compile-verified
//
#include <hip/hip_runtime.h>
#include <stdint.h>

// Problem constants (fixed by the reference setup_inputs()).
#define NBATCH  8
#define NSAMP   4194304          // samples per row
#define SEG     1024
#define HOP     512
#define NSEG    8191             // (NSAMP - SEG)/HOP + 1

#define THREADS 256
#define GROUPS  8                            // float4-groups per thread
#define CHUNK   (THREADS * 4 * GROUPS)       // 8192 samples per block (1024-aligned)

typedef float f4  __attribute__((ext_vector_type(4)));
typedef int   v4i __attribute__((ext_vector_type(4)));

typedef __attribute__((address_space(1))) v4i g_v4i;   // global int4
typedef __attribute__((address_space(3))) v4i l_v4i;   // LDS int4

// Address-space pointer helpers (integer detour: flat->AS cast via uintptr_t).
// Per CDNA5 ISA, flat shared addr[31:0] == LDS byte address, so the 32-bit
// truncation yields a valid LDS address for the async-copy VDST operand.
#define AS1_V4I(p) ((g_v4i*)(uintptr_t)(p))
#define AS3_V4I(p) ((l_v4i*)(uint32_t)(uintptr_t)(p))

#if __has_builtin(__builtin_amdgcn_global_load_async_to_lds_b128)
#define HAVE_ASYNC_LDS 1
#else
#define HAVE_ASYNC_LDS 0
#endif

__global__ __launch_bounds__(THREADS) void SegmenterOLA_kernel(
    const float* __restrict__ x,     // [NBATCH, NSAMP]
    const float* __restrict__ aw,    // [SEG]
    const float* __restrict__ sw,    // [SEG]
    float* __restrict__ Xout,        // [NBATCH, NSEG, SEG]
    float* __restrict__ xrec)        // [NBATCH, NSAMP]
{
    __shared__ __align__(16) float aw_l[SEG];
    __shared__ __align__(16) float sw_l[SEG];

    const int tid = threadIdx.x;
    const int j4  = tid * 4;                 // 0..1020, covers SEG exactly

    // ---- Stage window tables into LDS (async copy engine) ----
#if HAVE_ASYNC_LDS
    __builtin_amdgcn_global_load_async_to_lds_b128(
        AS1_V4I(aw + j4), AS3_V4I(&aw_l[j4]), 0, 0);
    __builtin_amdgcn_global_load_async_to_lds_b128(
        AS1_V4I(sw + j4), AS3_V4I(&sw_l[j4]), 0, 0);
#if __has_builtin(__builtin_amdgcn_s_wait_asynccnt)
    __builtin_amdgcn_s_wait_asynccnt(0);
#else
    asm volatile("s_wait_asynccnt 0x0" ::: "memory");
#endif
#else
    *(f4*)(&aw_l[j4]) = *(const f4*)(aw + j4);
    *(f4*)(&sw_l[j4]) = *(const f4*)(sw + j4);
#endif
    __syncthreads();

    // ---- Per-thread window values (invariant across the 8 groups) ----
    const int jl = j4 & (HOP - 1);           // 0..508, low-half window index
    const f4 a_lo = *(const f4*)(&aw_l[jl]);         // ds_load_b128
    const f4 s_lo = *(const f4*)(&sw_l[jl]);
    const f4 a_hi = *(const f4*)(&aw_l[jl + HOP]);
    const f4 s_hi = *(const f4*)(&sw_l[jl + HOP]);
    const f4 p_lo = a_lo * s_lo;                     // aw*sw products
    const f4 p_hi = a_hi * s_hi;

    const int blocksPerRow = NSAMP / CHUNK;          // 512
    const int b    = blockIdx.x / blocksPerRow;
    const int base = (blockIdx.x % blocksPerRow) * CHUNK;

    const size_t rowX = (size_t)b * NSEG * SEG;
    const size_t rowR = (size_t)b * NSAMP;

#pragma unroll
    for (int g = 0; g < GROUPS; ++g) {
        const int n  = base + g * SEG + j4;          // sample index in row
        const int s1 = n >> 9;                       // frame for low-half hit

        // Zero-reuse stream: non-temporal load, don't churn WGP$/L2.
        const f4 v = __builtin_nontemporal_load((const f4*)(x + rowR + n));

        if (g + 1 < GROUPS)                          // near-cache stream-ahead
            __builtin_prefetch(x + rowR + n + SEG, 0, 3);

        const bool c1 = (s1 <= NSEG - 1);            // frame s1 exists
        const bool c2 = (s1 >= 1);                   // frame s1-1 exists

        // X[b, s1, jl] and X[b, s1-1, jl+HOP]: each X element written once.
        if (c1) {
            f4 o = v * a_lo;
            __builtin_nontemporal_store(o, (f4*)(Xout + rowX + (size_t)s1 * SEG + jl));
        }
        if (c2) {
            f4 o = v * a_hi;
            __builtin_nontemporal_store(o, (f4*)(Xout + rowX + (size_t)(s1 - 1) * SEG + jl + HOP));
        }

        // Overlap-add collapses to a per-sample scale: no scatter, no atomics.
        f4 m = {0.f, 0.f, 0.f, 0.f};
        if (c1) m += p_lo;
        if (c2) m += p_hi;
        f4 r = v * m;
        __builtin_nontemporal_store(r, (f4*)(xrec + rowR + n));
    }
}

extern "C" void kernel_launch(void* const* d_in, const int* in_sizes, int n_in,
                              void* d_out, int out_size, void* d_ws, size_t ws_size,
                              hipStream_t stream) {
    const float* x  = (const float*)d_in[0];
    const float* aw = (const float*)d_in[1];
    const float* sw = (const float*)d_in[2];
    // d_out = [X (NBATCH*NSEG*SEG) | x_rec (NBATCH*NSAMP)] flat, in return order.
    float* Xout = (float*)d_out;
    float* xrec = (float*)d_out + (size_t)NBATCH * NSEG * SEG;

    const int grid = NBATCH * (NSAMP / CHUNK);       // 4096 blocks
    SegmenterOLA_kernel<<<grid, THREADS, 0, stream>>>(x, aw, sw, Xout, xrec);
}